// GIB_33809982554184
// MI455X (gfx1250) — compile-verified
//
#include <hip/hip_runtime.h>
#include <hip/hip_bf16.h>

// ---------------- problem constants (from reference) ----------------
constexpr int N_NODES = 16384;
constexpr int N_EDGES = 524288;
constexpr int F_IN    = 128;
constexpr int H       = 256;   // H1 == H2 == 256
constexpr int D1      = 64;

typedef float v2f __attribute__((ext_vector_type(2)));
typedef float v8f __attribute__((ext_vector_type(8)));

// ---------------- utility kernels ----------------
__global__ void fill_kernel(float* __restrict__ p, float v, int n) {
    int i = blockIdx.x * blockDim.x + threadIdx.x;
    if (i < n) p[i] = v;
}

// deg[d] += 1 for every edge (deg pre-initialized to 1.0 for self-loops)
__global__ void deg_kernel(const int* __restrict__ dst, float* __restrict__ deg) {
    int e = blockIdx.x * blockDim.x + threadIdx.x;
    if (e < N_EDGES) atomicAdd(&deg[dst[e]], 1.0f);
}

__global__ void rsqrt_kernel(float* __restrict__ p, int n) {
    int i = blockIdx.x * blockDim.x + threadIdx.x;
    if (i < n) p[i] = rsqrtf(p[i]);   // deg >= 1 always (self loop)
}

// ---------------- WMMA f32 GEMM: C[M,256] = A[M,K] x B[K,256] ----------------
// 4 waves per block; each wave owns a 16x32 output strip (2 accumulators share
// one A fragment). K templated -> full unroll -> loads fold to base+imm-offset,
// loop body is pure global_load + v_wmma.
template <int K>
__global__ __launch_bounds__(128)
void gemm_f32_wmma(const float* __restrict__ A, const float* __restrict__ B,
                   float* __restrict__ C) {
    constexpr int NC = 256;
    const int lane = threadIdx.x & 31;
    const int wave = threadIdx.x >> 5;
    const int half = lane >> 4;        // 0: lanes 0-15, 1: lanes 16-31
    const int l15  = lane & 15;

    const int tm = blockIdx.x * 16;                   // row tile
    const int tn = (blockIdx.y * 4 + wave) * 32;      // 32-wide col strip

    // Per-lane base pointers hoisted once; all loop accesses are +constant.
    const float* ap  = A + (size_t)(tm + l15) * K + 2 * half;  // A[row, k0+2*half]
    const float* bp0 = B + (size_t)(2 * half) * NC + tn + l15; // B[k0+2*half, n]
    const float* bp1 = bp0 + 16;

    v8f acc0 = {}, acc1 = {};
#pragma unroll
    for (int k0 = 0; k0 < K; k0 += 4) {
        float2 a2 = *(const float2*)(ap + k0);        // A frag: contiguous in K
        v2f av; av[0] = a2.x; av[1] = a2.y;
        v2f b0; b0[0] = bp0[k0 * NC]; b0[1] = bp0[k0 * NC + NC];
        v2f b1; b1[0] = bp1[k0 * NC]; b1[1] = bp1[k0 * NC + NC];

        acc0 = __builtin_amdgcn_wmma_f32_16x16x4_f32(
            false, av, false, b0, (short)0, acc0, false, false);
        acc1 = __builtin_amdgcn_wmma_f32_16x16x4_f32(
            false, av, false, b1, (short)0, acc1, false, false);
    }
    // D layout: VGPR r -> row (r + 8*half), col l15
#pragma unroll
    for (int r = 0; r < 8; ++r) {
        float* crow = C + (size_t)(tm + r + 8 * half) * NC + tn + l15;
        crow[0]  = acc0[r];
        crow[16] = acc1[r];
    }
}

// ---------------- edge scatter: out[dst] += x[src] * dinv[src]*dinv[dst] ----------------
__global__ void scatter_kernel(const int* __restrict__ src, const int* __restrict__ dst,
                               const float* __restrict__ x, const float* __restrict__ dinv,
                               float* __restrict__ out) {
    const int f = threadIdx.x;               // 256 threads = feature lane
    const int ebase = blockIdx.x * 4;
#pragma unroll
    for (int r = 0; r < 4; ++r) {
        const int e = ebase + r;
        const int s = src[e], d = dst[e];
        const float sc = dinv[s] * dinv[d];
        atomicAdd(&out[(size_t)d * H + f], x[(size_t)s * H + f] * sc);
    }
}

// out[i,f] = (maybe relu)( out[i,f] + xw[i,f]*dinv[i]^2 + b[f] )   (self-loop + bias)
__global__ void finalize_kernel(float* __restrict__ out, const float* __restrict__ xw,
                                const float* __restrict__ dinv, const float* __restrict__ b,
                                int relu) {
    const int idx = blockIdx.x * blockDim.x + threadIdx.x;
    const int i = idx >> 8, f = idx & 255;
    float dv = dinv[i];
    float v = out[idx] + xw[idx] * dv * dv + b[f];
    out[idx] = relu ? fmaxf(v, 0.0f) : v;
}

// ---------------- MLP head + group-feature accumulation ----------------
// 64 threads / block, 16 nodes / block; LDS-local gf accumulation, flushed once.
__global__ __launch_bounds__(64)
void head_kernel(const float* __restrict__ h2,
                 const float* __restrict__ fc1w, const float* __restrict__ fc1b,
                 const float* __restrict__ fc2w, const float* __restrict__ fc2b,
                 float* __restrict__ assign, float* __restrict__ gf) {
    __shared__ float row[H];
    __shared__ float a1[D1];
    __shared__ float asg[2];
    __shared__ float gfl[2 * H];
    const int t = threadIdx.x;

    for (int f = t; f < 2 * H; f += 64) gfl[f] = 0.0f;

    for (int nn = 0; nn < 16; ++nn) {
        const int i = blockIdx.x * 16 + nn;
        // 256 floats by 64 threads as float4: one b128 load each
        {
            const float4* src4 = (const float4*)(h2 + (size_t)i * H);
            ((float4*)row)[t] = src4[t];
        }
        __syncthreads();

        float acc = fc1b[t];
        const float4* w4 = (const float4*)(fc1w + (size_t)t * H);
        const float4* r4 = (const float4*)row;
#pragma unroll 8
        for (int k = 0; k < H / 4; ++k) {
            float4 a = r4[k], b = w4[k];
            acc += a.x * b.x + a.y * b.y + a.z * b.z + a.w * b.w;
        }
        a1[t] = tanhf(acc);
        __syncthreads();

        if (t < 2) {
            float l = fc2b[t];
            const float* w2 = fc2w + t * D1;
            for (int k = 0; k < D1; ++k) l += a1[k] * w2[k];
            asg[t] = l;
        }
        __syncthreads();
        if (t == 0) {
            float m = fmaxf(asg[0], asg[1]);
            float e0 = expf(asg[0] - m), e1 = expf(asg[1] - m);
            float inv = 1.0f / (e0 + e1);
            asg[0] = e0 * inv; asg[1] = e1 * inv;
        }
        __syncthreads();
        if (t < 2) assign[(size_t)i * 2 + t] = asg[t];

        const float s0 = asg[0], s1 = asg[1];
        for (int f = t; f < H; f += 64) {     // thread owns feature lanes: no race
            gfl[f]     += s0 * row[f];
            gfl[H + f] += s1 * row[f];
        }
        __syncthreads();
    }
    for (int f = t; f < 2 * H; f += 64) atomicAdd(&gf[f], gfl[f]);
}

// ---------------- new_adj = sum over edges of assign[src] (x) assign[dst] ----------------
__global__ void newadj_kernel(const int* __restrict__ src, const int* __restrict__ dst,
                              const float* __restrict__ assign, float* __restrict__ na) {
    __shared__ float s[4];
    if (threadIdx.x < 4) s[threadIdx.x] = 0.0f;
    __syncthreads();
    const float2* asg2 = (const float2*)assign;
    float l00 = 0, l01 = 0, l10 = 0, l11 = 0;
    const int base = blockIdx.x * blockDim.x * 4;
#pragma unroll
    for (int r = 0; r < 4; ++r) {
        const int e = base + r * blockDim.x + threadIdx.x;
        if (e < N_EDGES) {
            const float2 sa = asg2[src[e]];
            const float2 da = asg2[dst[e]];
            l00 += sa.x * da.x; l01 += sa.x * da.y;
            l10 += sa.y * da.x; l11 += sa.y * da.y;
        }
    }
    atomicAdd(&s[0], l00); atomicAdd(&s[1], l01);    // ds_add_f32 (LDS)
    atomicAdd(&s[2], l10); atomicAdd(&s[3], l11);
    __syncthreads();
    if (threadIdx.x < 4) atomicAdd(&na[threadIdx.x], s[threadIdx.x]);
}

// ---------------- final outputs ----------------
__global__ void final_kernel(const float* __restrict__ gf, const float* __restrict__ na,
                             float* __restrict__ out) {
    const int t = threadIdx.x;   // 256 threads
    const float g0 = gf[t], g1 = gf[H + t];
    out[t]         = 0.5f * (g0 + g1);                       // graph_embedding
    out[H + t]     = fminf(fmaxf(g0, -100.0f), 100.0f);      // positive
    out[2 * H + t] = fminf(fmaxf(g1, -100.0f), 100.0f);      // negative
    if (t == 0) {
        const float n00 = na[0], n01 = na[1], n10 = na[2], n11 = na[3];
        const float d0 = fmaxf(fabsf(n00) + fabsf(n01), 1e-12f);
        const float d1 = fmaxf(fabsf(n10) + fabsf(n11), 1e-12f);
        const float nd0 = n00 / d0 - 1.0f, nd1 = n11 / d1 - 1.0f;
        out[3 * H] = 0.5f * (nd0 * nd0 + nd1 * nd1);         // pos_penalty
    }
}

// ---------------- host orchestration ----------------
extern "C" void kernel_launch(void* const* d_in, const int* in_sizes, int n_in,
                              void* d_out, int out_size, void* d_ws, size_t ws_size,
                              hipStream_t stream) {
    const float* features = (const float*)d_in[0];
    const int*   edges    = (const int*)d_in[1];
    const int*   src      = edges;
    const int*   dst      = edges + N_EDGES;
    const float* W1  = (const float*)d_in[2];
    const float* b1  = (const float*)d_in[3];
    const float* W2  = (const float*)d_in[4];
    const float* b2  = (const float*)d_in[5];
    const float* fc1w = (const float*)d_in[6];
    const float* fc1b = (const float*)d_in[7];
    const float* fc2w = (const float*)d_in[8];
    const float* fc2b = (const float*)d_in[9];
    float* out = (float*)d_out;

    float* ws     = (float*)d_ws;
    float* bufA   = ws;                               // N*H  (GEMM output)
    float* bufB   = bufA + (size_t)N_NODES * H;       // N*H  (aggregation / h)
    float* dinv   = bufB + (size_t)N_NODES * H;       // N    (deg -> rsqrt)
    float* assign = dinv + N_NODES;                   // N*2
    float* gf     = assign + 2 * N_NODES;             // 2*H
    float* na     = gf + 2 * H;                       // 4   (contiguous after gf)

    const int NH = N_NODES * H;

    // degrees (self-loop = init 1.0), then dinv = rsqrt(deg)
    fill_kernel<<<N_NODES / 256, 256, 0, stream>>>(dinv, 1.0f, N_NODES);
    deg_kernel<<<N_EDGES / 256, 256, 0, stream>>>(dst, dinv);
    rsqrt_kernel<<<N_NODES / 256, 256, 0, stream>>>(dinv, N_NODES);

    // ---- layer 1: XW1 -> aggregate -> +self+bias, relu ----
    gemm_f32_wmma<F_IN><<<dim3(N_NODES / 16, H / 128), 128, 0, stream>>>(features, W1, bufA);
    fill_kernel<<<NH / 256, 256, 0, stream>>>(bufB, 0.0f, NH);
    scatter_kernel<<<N_EDGES / 4, 256, 0, stream>>>(src, dst, bufA, dinv, bufB);
    finalize_kernel<<<NH / 256, 256, 0, stream>>>(bufB, bufA, dinv, b1, 1);

    // ---- layer 2: h1@W2 -> aggregate -> +self+bias ----
    gemm_f32_wmma<H><<<dim3(N_NODES / 16, H / 128), 128, 0, stream>>>(bufB, W2, bufA);
    fill_kernel<<<NH / 256, 256, 0, stream>>>(bufB, 0.0f, NH);
    scatter_kernel<<<N_EDGES / 4, 256, 0, stream>>>(src, dst, bufA, dinv, bufB);
    finalize_kernel<<<NH / 256, 256, 0, stream>>>(bufB, bufA, dinv, b2, 0);   // bufB = h2

    // ---- head: assignment + group features; new_adj; outputs ----
    fill_kernel<<<3, 256, 0, stream>>>(gf, 0.0f, 2 * H + 4);   // gf and na contiguous
    head_kernel<<<N_NODES / 16, 64, 0, stream>>>(bufB, fc1w, fc1b, fc2w, fc2b, assign, gf);
    newadj_kernel<<<N_EDGES / (256 * 4), 256, 0, stream>>>(src, dst, assign, na);
    final_kernel<<<1, 256, 0, stream>>>(gf, na, out);
}